// DSnetwork_7249904796358
// MI455X (gfx1250) — compile-verified
//
#include <hip/hip_runtime.h>
#include <hip/hip_bf16.h>

typedef __attribute__((ext_vector_type(2))) float v2f;
typedef __attribute__((ext_vector_type(8))) float v8f;

#define EMB 128
#define EDGE_DIM 16
#define NUM_TASKS 10
#define BN_EPS 1e-5f

// ---------------------------------------------------------------------------
// Tiny sequential exclusive cumsum of num_subgraphs -> offsets[G+1]
// ---------------------------------------------------------------------------
__global__ void cumsum_kernel(const int* __restrict__ num_sub, int* __restrict__ offs, int G) {
    if (blockIdx.x == 0 && threadIdx.x == 0) {
        int a = 0;
        for (int i = 0; i < G; ++i) { offs[i] = a; a += num_sub[i]; }
        offs[G] = a;
    }
}

// ---------------------------------------------------------------------------
// eAgg[dst[e]][0:16] += edge_attr[e][0:16]   (4 lanes/edge, float4 each)
// ---------------------------------------------------------------------------
__global__ __launch_bounds__(256) void edge_attr_scatter(
    const int* __restrict__ dst, const float* __restrict__ edge_attr,
    float* __restrict__ eAgg, int E) {
    int gid = blockIdx.x * 256 + threadIdx.x;
    int e = gid >> 2;
    if (e >= E) return;
    int q = (gid & 3) * 4;
    int d = dst[e];
    const float4 v = *(const float4*)&edge_attr[(long)e * EDGE_DIM + q];
    float* p = &eAgg[(long)d * EDGE_DIM + q];
    unsafeAtomicAdd(p + 0, v.x);
    unsafeAtomicAdd(p + 1, v.y);
    unsafeAtomicAdd(p + 2, v.z);
    unsafeAtomicAdd(p + 3, v.w);
}

// ---------------------------------------------------------------------------
// aggH[dst[e]][:] += h[src[e]][:]   (32 lanes/edge, 4 floats per lane)
// ---------------------------------------------------------------------------
__global__ __launch_bounds__(256) void edge_scatter_h(
    const int* __restrict__ src, const int* __restrict__ dst,
    const float* __restrict__ h, float* __restrict__ agg, int E) {
    long gid = (long)blockIdx.x * 256 + threadIdx.x;
    int e = (int)(gid >> 5);
    if (e >= E) return;
    int lane = (int)(gid & 31) * 4;
    int s = src[e], d = dst[e];
    const float4 v = *(const float4*)&h[(long)s * EMB + lane];
    float* p = &agg[(long)d * EMB + lane];
    unsafeAtomicAdd(p + 0, v.x);
    unsafeAtomicAdd(p + 1, v.y);
    unsafeAtomicAdd(p + 2, v.z);
    unsafeAtomicAdd(p + 3, v.w);
}

// ---------------------------------------------------------------------------
// Fused GEMM via V_WMMA_F32_16X16X4_F32:
//   out = A1(Nx128)@W1(128x128) [+ A2(Nx128)@W2(128x128)] [+ A3(Nx16)@W3(16x128)] + bias
// Block: 256 threads = 8 wave32; block computes 32 rows x 128 cols.
// Wave w owns two 16x16 tiles (row halves) at column offset w*16, sharing
// every B fragment between the two accumulators (2 WMMA per B load).
// A-layout (ISA 7.12.2, 32-bit A 16x4): lane m=lane&15, hi=lane>>4;
//   vgpr0 = A[m][kb+2*hi], vgpr1 = A[m][kb+2*hi+1]
// C/D layout: vgpr v -> row (v + 8*hi), col = lane&15
// ---------------------------------------------------------------------------
__global__ __launch_bounds__(256) void gemm_fused_wmma(
    const float* __restrict__ A1, const float* __restrict__ W1,
    const float* __restrict__ A2, const float* __restrict__ W2,
    const float* __restrict__ A3, const float* __restrict__ W3,
    const float* __restrict__ bias, float* __restrict__ out, int nrows) {
    __shared__ float lA1[32][EMB];
    __shared__ float lA2[32][EMB];
    __shared__ float lA3[32][EDGE_DIM];

    const int t = threadIdx.x;
    const int row0 = blockIdx.x * 32;

    // --- stage A tiles into LDS (each thread: 16 floats per 32x128 matrix) ---
    {
        int r = t >> 3;             // 0..31
        int c = (t & 7) * 16;       // 0..112
        long gr = (long)(row0 + r);
        bool ok = (row0 + r) < nrows;
        float4 z4 = make_float4(0.f, 0.f, 0.f, 0.f);
        #pragma unroll
        for (int j = 0; j < 4; ++j) {
            float4 a = ok ? *(const float4*)&A1[gr * EMB + c + 4 * j] : z4;
            *(float4*)&lA1[r][c + 4 * j] = a;
        }
        if (A2) {
            #pragma unroll
            for (int j = 0; j < 4; ++j) {
                float4 a = ok ? *(const float4*)&A2[gr * EMB + c + 4 * j] : z4;
                *(float4*)&lA2[r][c + 4 * j] = a;
            }
        }
        if (A3) {
            // 256 threads -> 32x16: two elements per thread
            int r3 = t >> 3, c3 = (t & 7) * 2;
            lA3[r3][c3]     = ok ? A3[gr * EDGE_DIM + c3]     : 0.f;
            lA3[r3][c3 + 1] = ok ? A3[gr * EDGE_DIM + c3 + 1] : 0.f;
        }
    }
    __syncthreads();

    const int lane = t & 31;
    const int wave = t >> 5;
    const int n0 = wave * 16;
    const int m = lane & 15;
    const int hi = lane >> 4;
    const int n = lane & 15;

    v8f acc0 = {};   // rows row0 .. row0+15
    v8f acc1 = {};   // rows row0+16 .. row0+31

    // A1 @ W1 : K = 128
    #pragma unroll 4
    for (int kb = 0; kb < EMB; kb += 4) {
        int kk = kb + 2 * hi;
        v2f b; b.x = W1[(long)kk * EMB + n0 + n]; b.y = W1[(long)(kk + 1) * EMB + n0 + n];
        v2f a0; a0.x = lA1[m][kk];      a0.y = lA1[m][kk + 1];
        v2f a1; a1.x = lA1[m + 16][kk]; a1.y = lA1[m + 16][kk + 1];
        acc0 = __builtin_amdgcn_wmma_f32_16x16x4_f32(false, a0, false, b, (short)0, acc0, false, false);
        acc1 = __builtin_amdgcn_wmma_f32_16x16x4_f32(false, a1, false, b, (short)0, acc1, false, false);
    }
    // A2 @ W2 : K = 128
    if (A2) {
        #pragma unroll 4
        for (int kb = 0; kb < EMB; kb += 4) {
            int kk = kb + 2 * hi;
            v2f b; b.x = W2[(long)kk * EMB + n0 + n]; b.y = W2[(long)(kk + 1) * EMB + n0 + n];
            v2f a0; a0.x = lA2[m][kk];      a0.y = lA2[m][kk + 1];
            v2f a1; a1.x = lA2[m + 16][kk]; a1.y = lA2[m + 16][kk + 1];
            acc0 = __builtin_amdgcn_wmma_f32_16x16x4_f32(false, a0, false, b, (short)0, acc0, false, false);
            acc1 = __builtin_amdgcn_wmma_f32_16x16x4_f32(false, a1, false, b, (short)0, acc1, false, false);
        }
    }
    // A3 @ W3 : K = 16
    if (A3) {
        #pragma unroll
        for (int kb = 0; kb < EDGE_DIM; kb += 4) {
            int kk = kb + 2 * hi;
            v2f b; b.x = W3[(long)kk * EMB + n0 + n]; b.y = W3[(long)(kk + 1) * EMB + n0 + n];
            v2f a0; a0.x = lA3[m][kk];      a0.y = lA3[m][kk + 1];
            v2f a1; a1.x = lA3[m + 16][kk]; a1.y = lA3[m + 16][kk + 1];
            acc0 = __builtin_amdgcn_wmma_f32_16x16x4_f32(false, a0, false, b, (short)0, acc0, false, false);
            acc1 = __builtin_amdgcn_wmma_f32_16x16x4_f32(false, a1, false, b, (short)0, acc1, false, false);
        }
    }

    float bv = bias ? bias[n0 + n] : 0.f;
    #pragma unroll
    for (int v = 0; v < 8; ++v) {
        int r = row0 + v + 8 * hi;
        if (r < nrows) out[(long)r * EMB + n0 + n] = acc0[v] + bv;
        int r2 = r + 16;
        if (r2 < nrows) out[(long)r2 * EMB + n0 + n] = acc1[v] + bv;
    }
}

// ---------------------------------------------------------------------------
// BatchNorm statistics: per-feature sum / sumsq (block partials -> atomics)
// ---------------------------------------------------------------------------
__global__ __launch_bounds__(256) void bn_stats(
    const float* __restrict__ h1, float* __restrict__ bsum,
    float* __restrict__ bsq, int N) {
    int f = threadIdx.x & 127;
    int half = threadIdx.x >> 7;
    long r0 = (long)blockIdx.x * 64;
    float s = 0.f, q = 0.f;
    for (int i = half; i < 64; i += 2) {
        long r = r0 + i;
        if (r < N) { float v = h1[r * EMB + f]; s += v; q += v * v; }
    }
    __shared__ float ls[256], lq[256];
    ls[threadIdx.x] = s; lq[threadIdx.x] = q;
    __syncthreads();
    if (threadIdx.x < 128) {
        s = ls[threadIdx.x] + ls[threadIdx.x + 128];
        q = lq[threadIdx.x] + lq[threadIdx.x + 128];
        unsafeAtomicAdd(&bsum[f], s);
        unsafeAtomicAdd(&bsq[f], q);
    }
}

// ---------------------------------------------------------------------------
// BN apply + ReLU:  h = relu((h1-mu)*rsqrt(var+eps)*gamma + beta)
// ---------------------------------------------------------------------------
__global__ __launch_bounds__(256) void bn_apply(
    const float* __restrict__ h1, const float* __restrict__ bsum,
    const float* __restrict__ bsq, const float* __restrict__ gamma,
    const float* __restrict__ beta, float* __restrict__ h,
    long total, float invN) {
    long i = (long)blockIdx.x * 256 + threadIdx.x;
    if (i >= total) return;
    int f = (int)(i & 127);
    float mu = bsum[f] * invN;
    float var = bsq[f] * invN - mu * mu;
    float v = (h1[i] - mu) * rsqrtf(var + BN_EPS) * gamma[f] + beta[f];
    h[i] = fmaxf(v, 0.f);
}

// ---------------------------------------------------------------------------
// Node -> subgraph scatter (sum + count)
// ---------------------------------------------------------------------------
__global__ __launch_bounds__(256) void node_pool(
    const float* __restrict__ h, const int* __restrict__ subgraph_batch,
    const int* __restrict__ batch, const int* __restrict__ offs,
    float* __restrict__ sub_sum, float* __restrict__ sub_cnt, int N) {
    long gid = (long)blockIdx.x * 256 + threadIdx.x;
    int v = (int)(gid >> 5);
    if (v >= N) return;
    int lane = (int)(gid & 31) * 4;
    int sid = subgraph_batch[v] + offs[batch[v]];
    const float4 val = *(const float4*)&h[(long)v * EMB + lane];
    float* p = &sub_sum[(long)sid * EMB + lane];
    unsafeAtomicAdd(p + 0, val.x);
    unsafeAtomicAdd(p + 1, val.y);
    unsafeAtomicAdd(p + 2, val.z);
    unsafeAtomicAdd(p + 3, val.w);
    if ((gid & 31) == 0) unsafeAtomicAdd(&sub_cnt[sid], 1.f);
}

// ---------------------------------------------------------------------------
// Subgraph mean -> graph scatter (one wave per subgraph)
// ---------------------------------------------------------------------------
__global__ __launch_bounds__(256) void sub_pool(
    const float* __restrict__ sub_sum, const float* __restrict__ sub_cnt,
    const int* __restrict__ sub_id_batch,
    float* __restrict__ g_sum, float* __restrict__ g_cnt, int TOTAL_SUB) {
    int s = blockIdx.x * 8 + (threadIdx.x >> 5);
    if (s >= TOTAL_SUB) return;
    int lane = (threadIdx.x & 31) * 4;
    float c = fmaxf(sub_cnt[s], 1.f);
    float4 v = *(const float4*)&sub_sum[(long)s * EMB + lane];
    v.x /= c; v.y /= c; v.z /= c; v.w /= c;
    int g = sub_id_batch[s];
    float* p = &g_sum[(long)g * EMB + lane];
    unsafeAtomicAdd(p + 0, v.x);
    unsafeAtomicAdd(p + 1, v.y);
    unsafeAtomicAdd(p + 2, v.z);
    unsafeAtomicAdd(p + 3, v.w);
    if ((threadIdx.x & 31) == 0) unsafeAtomicAdd(&g_cnt[g], 1.f);
}

// ---------------------------------------------------------------------------
// Final MLP per graph: relu(hg@W1 + b1)@W2 + b2     (one block per graph)
// ---------------------------------------------------------------------------
__global__ __launch_bounds__(256) void final_mlp(
    const float* __restrict__ g_sum, const float* __restrict__ g_cnt,
    const float* __restrict__ W1, const float* __restrict__ b1,
    const float* __restrict__ W2, const float* __restrict__ b2,
    float* __restrict__ out) {
    __shared__ float hg[EMB];
    __shared__ float hid[2 * EMB];
    int g = blockIdx.x;
    int t = threadIdx.x;
    float c = fmaxf(g_cnt[g], 1.f);
    if (t < EMB) hg[t] = g_sum[(long)g * EMB + t] / c;
    __syncthreads();
    float a = b1[t];
    #pragma unroll 8
    for (int k = 0; k < EMB; ++k) a += hg[k] * W1[k * (2 * EMB) + t];
    hid[t] = fmaxf(a, 0.f);
    __syncthreads();
    if (t < NUM_TASKS) {
        float o = b2[t];
        #pragma unroll 8
        for (int k = 0; k < 2 * EMB; ++k) o += hid[k] * W2[k * NUM_TASKS + t];
        out[g * NUM_TASKS + t] = o;
    }
}

// ---------------------------------------------------------------------------
extern "C" void kernel_launch(void* const* d_in, const int* in_sizes, int n_in,
                              void* d_out, int out_size, void* d_ws, size_t ws_size,
                              hipStream_t stream) {
    const float* x            = (const float*)d_in[0];
    const int*   edge_index   = (const int*)d_in[1];
    const float* edge_attr    = (const float*)d_in[2];
    const int*   batch        = (const int*)d_in[3];
    const int*   sub_batch    = (const int*)d_in[4];
    const int*   sub_id_batch = (const int*)d_in[5];
    const int*   num_sub      = (const int*)d_in[6];
    const float* enc_W        = (const float*)d_in[7];
    const float* enc_b        = (const float*)d_in[8];
    const float* W_root       = (const float*)d_in[9];
    const float* b_root       = (const float*)d_in[10];
    const float* W_nbr        = (const float*)d_in[11];
    const float* W_edge       = (const float*)d_in[12];
    const float* bn_gamma     = (const float*)d_in[13];
    const float* bn_beta      = (const float*)d_in[14];
    const float* fin_W1       = (const float*)d_in[15];
    const float* fin_b1       = (const float*)d_in[16];
    const float* fin_W2       = (const float*)d_in[17];
    const float* fin_b2       = (const float*)d_in[18];

    const int N = in_sizes[0] / EMB;
    const int E = in_sizes[1] / 2;
    const int G = in_sizes[6];
    const int TOTAL_SUB = in_sizes[5];
    const int L = in_sizes[9] / (EMB * EMB);   // 3

    const int* e_src = edge_index;
    const int* e_dst = edge_index + E;

    // ---- workspace layout ----
    char* ws = (char*)d_ws;
    size_t off = 0;
    auto alloc = [&](size_t bytes) { char* p = ws + off; off += (bytes + 255) & ~(size_t)255; return p; };
    float* h        = (float*)alloc((size_t)N * EMB * sizeof(float));
    float* h1       = (float*)alloc((size_t)N * EMB * sizeof(float));
    float* aggH     = (float*)alloc((size_t)N * EMB * sizeof(float));
    float* eAgg     = (float*)alloc((size_t)N * EDGE_DIM * sizeof(float));
    int*   offs     = (int*)  alloc((size_t)(G + 1) * sizeof(int));
    float* bn_sum   = (float*)alloc(2 * EMB * sizeof(float));       // sum ++ sumsq
    float* sub_sum  = (float*)alloc((size_t)TOTAL_SUB * EMB * sizeof(float));
    float* sub_cnt  = (float*)alloc((size_t)TOTAL_SUB * sizeof(float));
    float* g_sum    = (float*)alloc((size_t)G * EMB * sizeof(float));
    float* g_cnt    = (float*)alloc((size_t)G * sizeof(float));
    (void)ws_size;

    // ---- zero accumulators (graph-capture-safe) ----
    hipMemsetAsync(eAgg, 0, (size_t)N * EDGE_DIM * sizeof(float), stream);
    hipMemsetAsync(sub_sum, 0, (size_t)TOTAL_SUB * EMB * sizeof(float), stream);
    hipMemsetAsync(sub_cnt, 0, (size_t)TOTAL_SUB * sizeof(float), stream);
    hipMemsetAsync(g_sum, 0, (size_t)G * EMB * sizeof(float), stream);
    hipMemsetAsync(g_cnt, 0, (size_t)G * sizeof(float), stream);

    cumsum_kernel<<<1, 32, 0, stream>>>(num_sub, offs, G);

    // segment_sum(edge_attr, dst) -- layer invariant
    {
        int blocks = (E * 4 + 255) / 256;
        edge_attr_scatter<<<blocks, 256, 0, stream>>>(e_dst, edge_attr, eAgg, E);
    }

    // encoder: h = x @ enc_W + enc_b
    {
        int blocks = (N + 31) / 32;
        gemm_fused_wmma<<<blocks, 256, 0, stream>>>(x, enc_W, nullptr, nullptr,
                                                    nullptr, nullptr, enc_b, h, N);
    }

    const long total = (long)N * EMB;
    const int bnApplyBlocks = (int)((total + 255) / 256);
    const float invN = 1.0f / (float)N;

    for (int l = 0; l < L; ++l) {
        hipMemsetAsync(aggH, 0, (size_t)N * EMB * sizeof(float), stream);
        {
            long work = (long)E * 32;
            int blocks = (int)((work + 255) / 256);
            edge_scatter_h<<<blocks, 256, 0, stream>>>(e_src, e_dst, h, aggH, E);
        }
        // h1 = h@W_root[l] + aggH@W_nbr[l] + eAgg@W_edge[l] + b_root[l]
        {
            int blocks = (N + 31) / 32;
            gemm_fused_wmma<<<blocks, 256, 0, stream>>>(
                h, W_root + (size_t)l * EMB * EMB,
                aggH, W_nbr + (size_t)l * EMB * EMB,
                eAgg, W_edge + (size_t)l * EDGE_DIM * EMB,
                b_root + (size_t)l * EMB, h1, N);
        }
        hipMemsetAsync(bn_sum, 0, 2 * EMB * sizeof(float), stream);
        {
            int blocks = (N + 63) / 64;
            bn_stats<<<blocks, 256, 0, stream>>>(h1, bn_sum, bn_sum + EMB, N);
        }
        bn_apply<<<bnApplyBlocks, 256, 0, stream>>>(
            h1, bn_sum, bn_sum + EMB,
            bn_gamma + (size_t)l * EMB, bn_beta + (size_t)l * EMB,
            h, total, invN);
    }

    // pooling
    {
        long work = (long)N * 32;
        int blocks = (int)((work + 255) / 256);
        node_pool<<<blocks, 256, 0, stream>>>(h, sub_batch, batch, offs,
                                              sub_sum, sub_cnt, N);
    }
    {
        int blocks = (TOTAL_SUB + 7) / 8;
        sub_pool<<<blocks, 256, 0, stream>>>(sub_sum, sub_cnt, sub_id_batch,
                                             g_sum, g_cnt, TOTAL_SUB);
    }
    final_mlp<<<G, 256, 0, stream>>>(g_sum, g_cnt, fin_W1, fin_b1,
                                     fin_W2, fin_b2, (float*)d_out);
}